// Qu_conv_mutual_6794638263033
// MI455X (gfx1250) — compile-verified
//
#include <hip/hip_runtime.h>

// ---------------------------------------------------------------------------
// MI455X / gfx1250 implementation.
// Phase 1: gather+gram via V_WMMA_F32_16X16X4_F32 (rank-4 outer-product steps),
//          branch-free (cndmask zeroing, uniform SALU loop), atomic merge.
// Phase 2: rot 2x2s, CNOT-ring permutation, P*L1 and L2^T entrywise.
// Phase 3: U = L2 * (P*L1) complex 256^3 matmul via f32 WMMA; A-panel staged
//          into LDS with the Tensor Data Mover (tensor_load_to_lds) and shared
//          by 8 waves per block.
// Phase 4: diag(U rho U^h) = rowwise <U.rho, U> via f32 WMMA; U^T panel via
//          TDM, rho kron factors via global_load_async_to_lds_b128.
// Phase 5: measure + conv2d + fc head, scalar.
// ---------------------------------------------------------------------------

typedef float v2f __attribute__((ext_vector_type(2)));
typedef float v8f __attribute__((ext_vector_type(8)));
typedef unsigned int v4u __attribute__((ext_vector_type(4)));
typedef int v4i __attribute__((ext_vector_type(4)));
typedef int v8i __attribute__((ext_vector_type(8)));

#define AS1 __attribute__((address_space(1)))
#define AS3 __attribute__((address_space(3)))

#if __has_builtin(__builtin_amdgcn_tensor_load_to_lds)
#define HAVE_TDM 1
#else
#define HAVE_TDM 0
#endif

#if __has_builtin(__builtin_amdgcn_global_load_async_to_lds_b128)
#define HAVE_ASYNC 1
#else
#define HAVE_ASYNC 0
#endif

__device__ inline v8f wmma4(v2f a, v2f b, v8f c) {
  // D(16x16,f32) += A(16x4,f32) * B(4x16,f32)
  return __builtin_amdgcn_wmma_f32_16x16x4_f32(
      false, a, false, b, (short)0, c, false, false);
}

__device__ inline unsigned lds_off(const void* p) {
  return (unsigned)(unsigned long long)(AS3 const void*)p;
}

__device__ inline void wait_asynccnt0() {
#if __has_builtin(__builtin_amdgcn_s_wait_asynccnt)
  __builtin_amdgcn_s_wait_asynccnt(0);
#else
  asm volatile("s_wait_asynccnt 0x0" ::: "memory");
#endif
}

#if HAVE_TDM
// TDM: load a 16-f32-wide x 256-row tile (row stride 256 f32) into contiguous
// LDS. D# group0: count=1, lds_addr, global_addr, type=2 ("image").
// D# group1: data_size=4B, tensor_dim0/1=256, tile_dim0=16, tile_dim1=256,
// tensor_dim0_stride=256. Groups 2/3 unused (tile_dim3/4 = 0).
__device__ inline void tdm_load_panel(const float* gsrc, unsigned lds_byte_off) {
  const unsigned long long ga = (unsigned long long)(size_t)gsrc;
  v4u g0;
  g0.x = 1u;                                                // count=1
  g0.y = lds_byte_off;                                      // lds_addr
  g0.z = (unsigned)ga;                                      // global_addr lo
  g0.w = (unsigned)((ga >> 32) & 0x1ffffffu) | (2u << 30);  // addr hi | type=2
  v8i g1;
  g1[0] = (int)(2u << 16);    // data_size = 2 (4 bytes)
  g1[1] = (int)(256u << 16);  // tensor_dim0[15:0] = 256
  g1[2] = (int)(256u << 16);  // tensor_dim0 hi | tensor_dim1[15:0] = 256
  g1[3] = (int)(16u << 16);   // tensor_dim1 hi | tile_dim0 = 16
  g1[4] = 256;                // tile_dim1 = 256, tile_dim2 = 0
  g1[5] = 256;                // tensor_dim0_stride = 256
  g1[6] = 0;
  g1[7] = 0;
  v4i gz4 = {0, 0, 0, 0};
  v8i gz8 = {0, 0, 0, 0, 0, 0, 0, 0};
  __builtin_amdgcn_tensor_load_to_lds(g0, g1, gz4, gz4, gz8, 0);
}
#endif

// workspace layout (float offsets)
constexpr int GRAM_D = 0;        // 256
constexpr int GRAM_T = 256;      // 256
constexpr int RHO_D  = 512;      // 256
constexpr int RHO_T  = 768;      // 256
constexpr int C_D    = 1024;     // 64
constexpr int C_T    = 1088;     // 64
constexpr int DIAG   = 1152;     // 512 (dt: 256, td: 256)
constexpr int PL1_RE = 2048;             // 256x256
constexpr int PL1_IM = PL1_RE + 65536;
constexpr int L2T_RE = PL1_IM + 65536;   // transposed: L2T[k][i] = L2[i][k]
constexpr int L2T_IM = L2T_RE + 65536;
constexpr int U_RE   = L2T_IM + 65536;   // row-major U
constexpr int U_IM   = U_RE   + 65536;
constexpr int UT_RE  = U_IM   + 65536;   // transposed U (WMMA A operand)
constexpr int UT_IM  = UT_RE  + 65536;

__device__ inline float2 cmul(float2 a, float2 b) {
  return make_float2(a.x * b.x - a.y * b.y, a.x * b.y + a.y * b.x);
}

// ---------------------------------------------------------------------------
__global__ void zero_kernel(float* p, int n) {
  int i = blockIdx.x * blockDim.x + threadIdx.x;
  if (i < n) p[i] = 0.0f;
}

// ---------------------------------------------------------------------------
// gram += e^T e over gathered embedding rows (padding_idx=0 -> zero row).
// One wave consumes 4 rows per WMMA: A = chunk^T (16x4), B = chunk (4x16);
// the f32 operand layouts make the A and B register images identical.
// Loop is SALU-uniform (readfirstlane'd wave id) and loads are branch-free
// (value cndmask for padding rows) so EXEC stays all-ones for WMMA.
__global__ void gram_accum_kernel(const int* __restrict__ idx, int n,
                                  const float* __restrict__ emb,
                                  float* __restrict__ gram_out) {
  const int lane = threadIdx.x & 31;
  const int wv   = __builtin_amdgcn_readfirstlane(threadIdx.x >> 5);
  const int wid  = blockIdx.x * (blockDim.x >> 5) + wv;
  const int nw   = gridDim.x * (blockDim.x >> 5);
  const int col  = lane & 15;
  const int kg   = lane >> 4;  // K half-group: 0 -> K{0,1}, 1 -> K{2,3}
  v8f acc = {};
  const int nchunk = n >> 2;
  for (int ch = wid; ch < nchunk; ch += nw) {
    const int r0 = (ch << 2) + (kg << 1);
    const int i0 = idx[r0];
    const int i1 = idx[r0 + 1];
    const float e0 = emb[i0 * 16 + col];  // row 0 is a valid address
    const float e1 = emb[i1 * 16 + col];
    v2f a;
    a.x = i0 ? e0 : 0.0f;  // v_cndmask, no EXEC change
    a.y = i1 ? e1 : 0.0f;
    acc = wmma4(a, a, acc);
  }
  // C layout: vgpr v, lane l -> element (M = v + 8*(l>>4), N = l&15)
  for (int v = 0; v < 8; ++v) {
    const int M = v + 8 * kg;
    atomicAdd(&gram_out[M * 16 + col], acc[v]);
  }
}

// ---------------------------------------------------------------------------
// conv1d -> c, rho = (c^T c)/tr, rotation 2x2s, CNOT permutation, P*L1, L2^T.
__global__ void setup_kernel(const float* __restrict__ dconv1_w,
                             const float* __restrict__ dconv1_b,
                             const float* __restrict__ tconv1_w,
                             const float* __restrict__ tconv1_b,
                             const float* __restrict__ mutual_w,
                             float* __restrict__ ws) {
  __shared__ float  s_gram[2][256];
  __shared__ float  s_c[2][64];
  __shared__ float  s_m[2][256];
  __shared__ float2 s_R[2][8][2][2];  // [layer][qubit][row][col]
  __shared__ int    s_inv[256];       // inverse CNOT-ring permutation
  const int t = threadIdx.x;  // 256 threads

  s_gram[0][t] = ws[GRAM_D + t];
  s_gram[1][t] = ws[GRAM_T + t];

  // CNOT ring: qubit q <-> bit (7-q) (qubit 0 = MSB). pi maps |x> -> |pi(x)>.
  {
    int y = t;
    for (int q = 0; q < 7; ++q)
      if ((y >> (7 - q)) & 1) y ^= 1 << (6 - q);
    if (y & 1) y ^= 0x80;  // CNOT(7,0)
    s_inv[y] = t;          // pi^{-1}
  }

  // Rz(tz) Ry(ty) Rx(tx) per qubit, per layer
  if (t < 16) {
    const int q = t & 7, layer = t >> 3;
    const float WMUL = 0.63245553203367586640f;  // sqrt(2/5)
    const float tx = mutual_w[q * 6 + layer * 3 + 0] * WMUL;
    const float ty = mutual_w[q * 6 + layer * 3 + 1] * WMUL;
    const float tz = mutual_w[q * 6 + layer * 3 + 2] * WMUL;
    const float hx = 0.5f * tx, hy = 0.5f * ty, hz = 0.5f * tz;
    const float cx = cosf(hx), sx = sinf(hx);
    const float cy = cosf(hy), sy = sinf(hy);
    const float cz = cosf(hz), sz = sinf(hz);
    // M = Ry @ Rx
    const float2 M00 = make_float2( cy * cx,  sy * sx);
    const float2 M01 = make_float2(-sy * cx, -cy * sx);
    const float2 M10 = make_float2( sy * cx, -cy * sx);
    const float2 M11 = make_float2( cy * cx, -sy * sx);
    const float2 e0 = make_float2(cz, -sz);  // row0 phase
    const float2 e1 = make_float2(cz,  sz);  // row1 phase
    s_R[layer][q][0][0] = cmul(M00, e0);
    s_R[layer][q][0][1] = cmul(M01, e0);
    s_R[layer][q][1][0] = cmul(M10, e1);
    s_R[layer][q][1][1] = cmul(M11, e1);
  }
  __syncthreads();

  // conv1d 'same' k=4 pad (1,2): c[o][j]
  if (t < 128) {
    const int br = t >> 6, o = (t >> 4) & 3, j = t & 15;
    const float* w = br ? tconv1_w : dconv1_w;
    const float* b = br ? tconv1_b : dconv1_b;
    float acc = b[o];
    for (int i = 0; i < 16; ++i)
      for (int k = 0; k < 4; ++k) {
        const int jj = j + k - 1;
        if (jj >= 0 && jj < 16)
          acc += s_gram[br][i * 16 + jj] * w[(o * 16 + i) * 4 + k];
      }
    s_c[br][o * 16 + j] = acc;
    ws[(br ? C_T : C_D) + o * 16 + j] = acc;
  }
  __syncthreads();

  // m = c^T c
  for (int e = t; e < 512; e += 256) {
    const int br = e >> 8, r = (e >> 4) & 15, cc = e & 15;
    float acc = 0.0f;
    for (int o = 0; o < 4; ++o)
      acc += s_c[br][o * 16 + r] * s_c[br][o * 16 + cc];
    s_m[br][r * 16 + cc] = acc;
  }
  __syncthreads();

  // rho = m / trace(m)
  for (int e = t; e < 512; e += 256) {
    const int br = e >> 8, i = e & 255;
    float tr = 0.0f;
    for (int r = 0; r < 16; ++r) tr += s_m[br][r * 17];
    ws[(br ? RHO_T : RHO_D) + i] = s_m[br][i] / tr;
  }

  // PL1[i][j] = L1[pi^{-1}(i)][j];  L2T[k][i] = L2[i][k]
  {
    const int src = s_inv[t];
    for (int j = 0; j < 256; ++j) {
      float re = 1.0f, im = 0.0f;
      for (int q = 0; q < 8; ++q) {
        const float2 rq = s_R[0][q][(src >> (7 - q)) & 1][(j >> (7 - q)) & 1];
        const float nre = re * rq.x - im * rq.y;
        im = re * rq.y + im * rq.x;
        re = nre;
      }
      ws[PL1_RE + t * 256 + j] = re;
      ws[PL1_IM + t * 256 + j] = im;

      re = 1.0f; im = 0.0f;
      for (int q = 0; q < 8; ++q) {
        const float2 rq = s_R[1][q][(j >> (7 - q)) & 1][(t >> (7 - q)) & 1];
        const float nre = re * rq.x - im * rq.y;
        im = re * rq.y + im * rq.x;
        re = nre;
      }
      ws[L2T_RE + t * 256 + j] = re;
      ws[L2T_IM + t * 256 + j] = im;
    }
  }
}

// ---------------------------------------------------------------------------
// U = L2 @ PL1 (complex 256x256). 16 blocks (one per 16-row tile band) x 8
// waves; the A panel (L2^T columns for this band, 16x256 re+im) is staged into
// LDS once via TDM and reused by all 8 waves (2 output tiles each).
__global__ void __launch_bounds__(256) umat_kernel(float* __restrict__ ws) {
  __shared__ __align__(16) float sA_re[256 * 16];
  __shared__ __align__(16) float sA_im[256 * 16];
  const int ti   = blockIdx.x;  // tile-row band
  const int lane = threadIdx.x & 31;
  const int col  = lane & 15, kg = lane >> 4;

#if HAVE_TDM
  if (threadIdx.x < 32) {
    tdm_load_panel(ws + L2T_RE + ti * 16, lds_off(sA_re));
    tdm_load_panel(ws + L2T_IM + ti * 16, lds_off(sA_im));
    __builtin_amdgcn_s_wait_tensorcnt(0);
  }
#else
  for (int i = threadIdx.x; i < 4096; i += 256) {
    sA_re[i] = ws[L2T_RE + (i >> 4) * 256 + ti * 16 + (i & 15)];
    sA_im[i] = ws[L2T_IM + (i >> 4) * 256 + ti * 16 + (i & 15)];
  }
#endif
  __syncthreads();
  asm volatile("" ::: "memory");

  const int wv = __builtin_amdgcn_readfirstlane(threadIdx.x >> 5);
  const float* p1re = ws + PL1_RE;
  const float* p1im = ws + PL1_IM;
  for (int tt = 0; tt < 2; ++tt) {
    const int tj = wv * 2 + tt;
    v8f rr = {}, ii = {}, ri = {}, ir = {};
    for (int k0 = 0; k0 < 256; k0 += 4) {
      const int ka = k0 + (kg << 1);
      v2f are, aim, bre, bim;
      // A[m,k] = L2[ti*16+m][k], staged in LDS at float offset k*16 + m
      are.x = sA_re[ka * 16 + col];
      are.y = sA_re[(ka + 1) * 16 + col];
      aim.x = sA_im[ka * 16 + col];
      aim.y = sA_im[(ka + 1) * 16 + col];
      // B[k,n] = PL1[k][tj*16+n]
      bre.x = p1re[ka * 256 + tj * 16 + col];
      bre.y = p1re[(ka + 1) * 256 + tj * 16 + col];
      bim.x = p1im[ka * 256 + tj * 16 + col];
      bim.y = p1im[(ka + 1) * 256 + tj * 16 + col];
      rr = wmma4(are, bre, rr);
      ii = wmma4(aim, bim, ii);
      ri = wmma4(are, bim, ri);
      ir = wmma4(aim, bre, ir);
    }
    for (int v = 0; v < 8; ++v) {
      const int M = v + 8 * kg;
      const int gi = ti * 16 + M, gj = tj * 16 + col;
      const float ure = rr[v] - ii[v];
      const float uim = ri[v] + ir[v];
      ws[U_RE + gi * 256 + gj] = ure;
      ws[U_IM + gi * 256 + gj] = uim;
      ws[UT_RE + gj * 256 + gi] = ure;
      ws[UT_IM + gj * 256 + gi] = uim;
    }
  }
}

// ---------------------------------------------------------------------------
// diag(U rho U^h) for rho = kron(rA, rB): V = U * rho via WMMA (rho generated
// on the fly from 16x16 factors in LDS), then diag[n] = sum_a ReV*ReU+ImV*ImU.
// Grid: 2 orderings x 16 row-tiles, one wave each. The U^T A-panel is staged
// into LDS via TDM (reused across the 16 column tiles); rho factors arrive via
// async global->LDS b128 copies.
__global__ void __launch_bounds__(32) sandwich_diag_kernel(float* __restrict__ ws) {
  __shared__ __align__(16) float sA_re[256 * 16];
  __shared__ __align__(16) float sA_im[256 * 16];
  __shared__ __align__(16) float rA[256];
  __shared__ __align__(16) float rB[256];
  const int s    = blockIdx.x >> 4;   // 0: kron(d,t), 1: kron(t,d)
  const int rt   = blockIdx.x & 15;   // row tile
  const int lane = threadIdx.x & 31;
  const int col  = lane & 15, kg = lane >> 4;

  const float* pa = ws + (s ? RHO_T : RHO_D);
  const float* pb = ws + (s ? RHO_D : RHO_T);
#if HAVE_ASYNC
  {
    const int i4 = lane * 4;  // 4 floats (b128) per lane
    __builtin_amdgcn_global_load_async_to_lds_b128(
        (AS1 v4i*)(pa + i4), (AS3 v4i*)&rA[i4], 0, 0);
    __builtin_amdgcn_global_load_async_to_lds_b128(
        (AS1 v4i*)(pa + 128 + i4), (AS3 v4i*)&rA[128 + i4], 0, 0);
    __builtin_amdgcn_global_load_async_to_lds_b128(
        (AS1 v4i*)(pb + i4), (AS3 v4i*)&rB[i4], 0, 0);
    __builtin_amdgcn_global_load_async_to_lds_b128(
        (AS1 v4i*)(pb + 128 + i4), (AS3 v4i*)&rB[128 + i4], 0, 0);
  }
#else
  for (int i = lane; i < 256; i += 32) { rA[i] = pa[i]; rB[i] = pb[i]; }
#endif
#if HAVE_TDM
  tdm_load_panel(ws + UT_RE + rt * 16, lds_off(sA_re));
  tdm_load_panel(ws + UT_IM + rt * 16, lds_off(sA_im));
  __builtin_amdgcn_s_wait_tensorcnt(0);
#else
  for (int i = lane; i < 4096; i += 32) {
    sA_re[i] = ws[UT_RE + (i >> 4) * 256 + rt * 16 + (i & 15)];
    sA_im[i] = ws[UT_IM + (i >> 4) * 256 + rt * 16 + (i & 15)];
  }
#endif
#if HAVE_ASYNC
  wait_asynccnt0();
#endif
  __syncthreads();
  asm volatile("" ::: "memory");

  const float* ure = ws + U_RE;
  const float* uim = ws + U_IM;
  float part[8];
  for (int v = 0; v < 8; ++v) part[v] = 0.0f;

  for (int cj = 0; cj < 16; ++cj) {
    v8f vre = {}, vim = {};
    const int c = cj * 16 + col;
    for (int k0 = 0; k0 < 256; k0 += 4) {
      const int ka = k0 + (kg << 1);
      v2f a_re, a_im, b;
      a_re.x = sA_re[ka * 16 + col];
      a_re.y = sA_re[(ka + 1) * 16 + col];
      a_im.x = sA_im[ka * 16 + col];
      a_im.y = sA_im[(ka + 1) * 16 + col];
      b.x = rA[(ka >> 4) * 16 + (c >> 4)] * rB[(ka & 15) * 16 + (c & 15)];
      b.y = rA[((ka + 1) >> 4) * 16 + (c >> 4)] * rB[((ka + 1) & 15) * 16 + (c & 15)];
      vre = wmma4(a_re, b, vre);
      vim = wmma4(a_im, b, vim);
    }
    for (int v = 0; v < 8; ++v) {
      const int M = v + 8 * kg;
      const int gi = rt * 16 + M, gj = cj * 16 + col;
      part[v] += vre[v] * ure[gi * 256 + gj] + vim[v] * uim[gi * 256 + gj];
    }
  }
  // reduce across 16 lanes carrying the same row
  __shared__ float red[32][8];
  for (int v = 0; v < 8; ++v) red[lane][v] = part[v];
  __syncthreads();
  if (lane < 16) {
    const int r = lane;               // rows 0..15
    const int g = r >> 3, v = r & 7;  // lane group / vgpr holding this row
    float sum = 0.0f;
    for (int c2 = 0; c2 < 16; ++c2) sum += red[g * 16 + c2][v];
    ws[DIAG + s * 256 + rt * 16 + r] = sum;
  }
}

// ---------------------------------------------------------------------------
// measure + conv2d + fc head.
__global__ void head_kernel(const float* __restrict__ dconv2_w,
                            const float* __restrict__ dconv2_b,
                            const float* __restrict__ tconv2_w,
                            const float* __restrict__ tconv2_b,
                            const float* __restrict__ fc1_w,
                            const float* __restrict__ fc1_b,
                            const float* __restrict__ fc2_w,
                            const float* __restrict__ fc2_b,
                            const float* __restrict__ ws,
                            float* __restrict__ out) {
  __shared__ float s_red[2][16];
  __shared__ float s_mv[2][4];
  __shared__ float s_pre[2][16];
  __shared__ float s_outer[2][16];
  __shared__ float s_x[128];
  __shared__ float s_h[32];
  const int t = threadIdx.x;  // 128 threads

  // ptrace diagonal: reduced[j] = sum_i diag[i*16 + j]
  if (t < 32) {
    const int s = t >> 4, j = t & 15;
    float acc = 0.0f;
    for (int i = 0; i < 16; ++i) acc += ws[DIAG + s * 256 + i * 16 + j];
    s_red[s][j] = acc;
  }
  __syncthreads();
  // Pauli-Z expectations: m[i] = sum_k sign(i,k) * reduced[k]
  if (t < 8) {
    const int s = t >> 2, i = t & 3;
    float acc = 0.0f;
    for (int k = 0; k < 16; ++k) {
      const float sign = ((k >> (3 - i)) & 1) ? -1.0f : 1.0f;
      acc += sign * s_red[s][k];
    }
    s_mv[s][i] = acc;
  }
  __syncthreads();
  // pre = c c^T (4x4), outer = m m^T (4x4)
  if (t < 32) {
    const int s = t >> 4, a = (t >> 2) & 3, b = t & 3;
    const float* c = ws + (s ? C_T : C_D);
    float acc = 0.0f;
    for (int j = 0; j < 16; ++j) acc += c[a * 16 + j] * c[b * 16 + j];
    s_pre[s][a * 4 + b] = acc;
    s_outer[s][a * 4 + b] = s_mv[s][a] * s_mv[s][b];
  }
  __syncthreads();
  // conv2d 3x3 pad 1, in=2ch, out=4ch, 4x4 spatial: 128 outputs
  {
    const int br = t >> 6, rem = t & 63;
    const int oc = rem >> 4, y = (rem >> 2) & 3, x = rem & 3;
    const float* w  = br ? tconv2_w : dconv2_w;
    const float* bb = br ? tconv2_b : dconv2_b;
    float acc = bb[oc];
    for (int ic = 0; ic < 2; ++ic) {
      const float* in = ic ? s_outer[br] : s_pre[br];
      for (int ky = 0; ky < 3; ++ky)
        for (int kx = 0; kx < 3; ++kx) {
          const int yy = y + ky - 1, xx = x + kx - 1;
          if (yy >= 0 && yy < 4 && xx >= 0 && xx < 4)
            acc += in[yy * 4 + xx] * w[((oc * 2 + ic) * 3 + ky) * 3 + kx];
        }
    }
    s_x[br * 64 + rem] = acc;
  }
  __syncthreads();
  if (t < 32) {
    float acc = fc1_b[t];
    for (int j = 0; j < 128; ++j) acc += s_x[j] * fc1_w[t * 128 + j];
    s_h[t] = acc > 0.0f ? acc : 0.01f * acc;
  }
  __syncthreads();
  if (t == 0) {
    float acc = fc2_b[0];
    for (int j = 0; j < 32; ++j) acc += s_h[j] * fc2_w[j];
    out[0] = acc > 0.0f ? acc : 0.01f * acc;
  }
}

// ---------------------------------------------------------------------------
extern "C" void kernel_launch(void* const* d_in, const int* in_sizes, int n_in,
                              void* d_out, int out_size, void* d_ws, size_t ws_size,
                              hipStream_t stream) {
  const int*   drug   = (const int*)d_in[0];
  const int*   target = (const int*)d_in[1];
  const float* emb_d  = (const float*)d_in[2];
  const float* emb_t  = (const float*)d_in[3];
  const float* dc1w   = (const float*)d_in[4];
  const float* dc1b   = (const float*)d_in[5];
  const float* tc1w   = (const float*)d_in[6];
  const float* tc1b   = (const float*)d_in[7];
  const float* dc2w   = (const float*)d_in[8];
  const float* dc2b   = (const float*)d_in[9];
  const float* tc2w   = (const float*)d_in[10];
  const float* tc2b   = (const float*)d_in[11];
  const float* mutual = (const float*)d_in[12];
  const float* fc1w   = (const float*)d_in[13];
  const float* fc1b   = (const float*)d_in[14];
  const float* fc2w   = (const float*)d_in[15];
  const float* fc2b   = (const float*)d_in[16];
  float* ws  = (float*)d_ws;
  float* out = (float*)d_out;

  const int nD = in_sizes[0];
  const int nT = in_sizes[1];

  zero_kernel<<<2, 256, 0, stream>>>(ws, 512);
  gram_accum_kernel<<<512, 256, 0, stream>>>(drug, nD, emb_d, ws + GRAM_D);
  gram_accum_kernel<<<512, 256, 0, stream>>>(target, nT, emb_t, ws + GRAM_T);
  setup_kernel<<<1, 256, 0, stream>>>(dc1w, dc1b, tc1w, tc1b, mutual, ws);
  umat_kernel<<<16, 256, 0, stream>>>(ws);
  sandwich_diag_kernel<<<32, 32, 0, stream>>>(ws);
  head_kernel<<<1, 128, 0, stream>>>(dc2w, dc2b, tc2w, tc2b,
                                     fc1w, fc1b, fc2w, fc2b, ws, out);
}